// WaveletAttention_3015067042172
// MI455X (gfx1250) — compile-verified
//
#include <hip/hip_runtime.h>
#include <hip/hip_bf16.h>
#include <math.h>

// Problem: B=16, C=256, H=W=160.  out = x * softmax(mean_hw(x) @ fc_w^T + fc_b)
// (wavelet terms cancel exactly in the global average pool -- see analysis).

typedef __attribute__((ext_vector_type(2))) float v2f;
typedef __attribute__((ext_vector_type(8))) float v8f;

#define PLANE 25600          // 160*160
#define NBC   4096           // 16*256
#define NB    16
#define NC    256

// ---------------------------------------------------------------------------
// Kernel 1: gap[b*256+c] = mean over 160x160 plane of x.
// One block per (b,c) plane; 256 threads * 25 float4 each = 25600 floats.
// ---------------------------------------------------------------------------
__global__ void __launch_bounds__(256)
wa_gap_kernel(const float* __restrict__ x, float* __restrict__ gap) {
    const int bc = blockIdx.x;
    const int t  = threadIdx.x;
    const float4* __restrict__ p = (const float4*)(x + (size_t)bc * PLANE);

    float s = 0.0f;
#pragma unroll
    for (int i = 0; i < 25; ++i) {
        float4 v = p[i * 256 + t];
        s += (v.x + v.y) + (v.z + v.w);
    }
    // intra-wave (wave32) reduction
#pragma unroll
    for (int off = 16; off > 0; off >>= 1)
        s += __shfl_down(s, off, 32);

    __shared__ float red[8];
    const int lane = t & 31;
    const int wv   = t >> 5;
    if (lane == 0) red[wv] = s;
    __syncthreads();
    if (t == 0) {
        float v = red[0];
#pragma unroll
        for (int i = 1; i < 8; ++i) v += red[i];
        gap[bc] = v * (1.0f / (float)PLANE);
    }
}

// ---------------------------------------------------------------------------
// Kernel 2: logits = gap(16x256) @ fc_w(256x256)^T + fc_b ; attn = softmax rows.
// Single block, 256 threads = 8 wave32s. Each wave computes 2 output N-tiles
// with V_WMMA_F32_16X16X4_F32 (64 K-steps of 4), results staged in LDS, then
// per-row softmax over the 256 channels.
//
// A-fragment (16x4 f32, per ISA 7.12.2): lane<16 -> {A[lane][k], A[lane][k+1]},
// lane>=16 -> {A[lane-16][k+2], A[lane-16][k+3]}.  B-fragment symmetric with
// B[k][col] = fc_w[col*256+k].  D/C: VGPR j, lane L -> (M = j+8*(L>=16), N=L&15).
// ---------------------------------------------------------------------------
__global__ void __launch_bounds__(256)
wa_fc_softmax_kernel(const float* __restrict__ gap,
                     const float* __restrict__ fc_w,
                     const float* __restrict__ fc_b,
                     float* __restrict__ attn) {
    __shared__ float logits[NB * NC];   // 16 KB

    const int lane = threadIdx.x & 31;
    const int wv   = threadIdx.x >> 5;

    const int row  = lane & 15;         // M index (batch)
    const int koff = (lane >> 4) << 1;  // 0 for lanes 0-15, 2 for lanes 16-31

    const float2* __restrict__ gp = (const float2*)gap;
    const float2* __restrict__ wp = (const float2*)fc_w;

    // each wave owns N-tiles {wv, wv+8}; uniform loop -> EXEC all ones
#pragma unroll
    for (int nt = 0; nt < 2; ++nt) {
        const int n   = wv + nt * 8;
        const int col = n * 16 + row;   // output channel for this lane's B column

        v8f acc = {0.f, 0.f, 0.f, 0.f, 0.f, 0.f, 0.f, 0.f};
#pragma unroll 4
        for (int k = 0; k < NC; k += 4) {
            float2 af = gp[(row * NC + k + koff) >> 1];
            float2 bf = wp[(col * NC + k + koff) >> 1];
            v2f a; a[0] = af.x; a[1] = af.y;
            v2f b; b[0] = bf.x; b[1] = bf.y;
            acc = __builtin_amdgcn_wmma_f32_16x16x4_f32(
                false, a, false, b, (short)0, acc, false, false);
        }
        const float bias = fc_b[col];
        const int mbase  = (lane >> 4) << 3;   // 0 or 8
#pragma unroll
        for (int j = 0; j < 8; ++j)
            logits[(mbase + j) * NC + col] = acc[j] + bias;
    }
    __syncthreads();

    // softmax: each wave handles rows {wv, wv+8}; 8 channels per lane
#pragma unroll
    for (int rb = 0; rb < 2; ++rb) {
        const int b = wv + rb * 8;
        float vals[8];
        float mx = -INFINITY;
#pragma unroll
        for (int i = 0; i < 8; ++i) {
            vals[i] = logits[b * NC + i * 32 + lane];
            mx = fmaxf(mx, vals[i]);
        }
#pragma unroll
        for (int off = 16; off > 0; off >>= 1)
            mx = fmaxf(mx, __shfl_xor(mx, off, 32));
        float sum = 0.0f;
#pragma unroll
        for (int i = 0; i < 8; ++i) {
            vals[i] = expf(vals[i] - mx);
            sum += vals[i];
        }
#pragma unroll
        for (int off = 16; off > 0; off >>= 1)
            sum += __shfl_xor(sum, off, 32);
        const float inv = 1.0f / sum;
#pragma unroll
        for (int i = 0; i < 8; ++i)
            attn[b * NC + i * 32 + lane] = vals[i] * inv;
    }
}

// ---------------------------------------------------------------------------
// Kernel 3: out[b,c,:,:] = x[b,c,:,:] * attn[b,c].  Pure bandwidth, float4.
// ---------------------------------------------------------------------------
__global__ void __launch_bounds__(256)
wa_scale_kernel(const float* __restrict__ x,
                const float* __restrict__ attn,
                float* __restrict__ out) {
    const int bc = blockIdx.x;
    const int t  = threadIdx.x;
    const float a = attn[bc];
    const float4* __restrict__ xp = (const float4*)(x   + (size_t)bc * PLANE);
    float4*       __restrict__ op = (float4*)      (out + (size_t)bc * PLANE);
#pragma unroll
    for (int i = 0; i < 25; ++i) {
        float4 v = xp[i * 256 + t];
        v.x *= a; v.y *= a; v.z *= a; v.w *= a;
        op[i * 256 + t] = v;
    }
}

// ---------------------------------------------------------------------------
extern "C" void kernel_launch(void* const* d_in, const int* in_sizes, int n_in,
                              void* d_out, int out_size, void* d_ws, size_t ws_size,
                              hipStream_t stream) {
    const float* x     = (const float*)d_in[0];   // (16,256,160,160)
    // d_in[1] = theta  (unused: cancels in GAP)
    // d_in[2] = fuse_w (unused: cancels in GAP)
    const float* fc_w  = (const float*)d_in[3];   // (256,256)
    const float* fc_b  = (const float*)d_in[4];   // (256,)
    float*       out   = (float*)d_out;

    float* gap  = (float*)d_ws;          // 4096 floats
    float* attn = gap + NBC;             // 4096 floats

    wa_gap_kernel<<<NBC, 256, 0, stream>>>(x, gap);
    wa_fc_softmax_kernel<<<1, 256, 0, stream>>>(gap, fc_w, fc_b, attn);
    wa_scale_kernel<<<NBC, 256, 0, stream>>>(x, attn, out);
}